// get_model_10857677324944
// MI455X (gfx1250) — compile-verified
//
#include <hip/hip_runtime.h>
#include <float.h>
#include <math.h>

// ---------------------------------------------------------------------------
// VN-PointNet encoder for MI455X (gfx1250, wave32).
// GEMM layers use V_WMMA_F32_16X16X4_F32 with fully padded operands (no
// guards in the hot loop). Large GEMMs use a 2x2 register-blocked kernel
// (32x32 output per wave): 1.5 load-instrs per WMMA + 4-way WMMA ILP.
// Workspace requirement: ~160 MB.
// ---------------------------------------------------------------------------

static constexpr int B_  = 8;
static constexpr int N_  = 2048;
static constexpr int KNN = 40;
static constexpr int C1 = 21, C2 = 42, C3 = 341, F1 = 170, F2 = 85;
static constexpr int Cp1 = 32, Cp2 = 48, Cp3 = 352;   // roundup16 channel pads
static constexpr int LCOLS = 3 * N_;                  // 6144 columns per batch
static constexpr float EPS_   = 1e-6f;
static constexpr float BNEPS_ = 1e-5f;

typedef __attribute__((ext_vector_type(2))) float v2f;
typedef __attribute__((ext_vector_type(8))) float v8f;

// --- feature build: [nbr-ctr, ctr, cross(nbr,ctr)] -------------------------
__device__ __forceinline__ void load_feat(const float* __restrict__ xb,
                                          int n, int nb, float f[3][3]) {
  float cx = xb[n],  cy = xb[N_ + n],  cz = xb[2 * N_ + n];
  float nx = xb[nb], ny = xb[N_ + nb], nz = xb[2 * N_ + nb];
  f[0][0] = nx - cx; f[0][1] = ny - cy; f[0][2] = nz - cz;
  f[1][0] = cx;      f[1][1] = cy;      f[1][2] = cz;
  f[2][0] = ny * cz - nz * cy;
  f[2][1] = nz * cx - nx * cz;
  f[2][2] = nx * cy - ny * cx;
}

// --- zero-padded weight copy: Wp(Mp x Kp) <- W(M x K) ----------------------
__global__ void pad_weight_kernel(const float* __restrict__ W, float* __restrict__ Wp,
                                  int M, int K, int Mp, int Kp) {
  int t = blockIdx.x * blockDim.x + threadIdx.x;
  if (t >= Mp * Kp) return;
  int m = t / Kp, k = t - m * Kp;
  Wp[t] = (m < M && k < K) ? W[m * K + k] : 0.f;
}

// --- kNN: one thread per point, maintain top-KNN by squared distance -------
__global__ void knn_kernel(const float* __restrict__ x, int* __restrict__ idx) {
  int t = blockIdx.x * blockDim.x + threadIdx.x;
  if (t >= B_ * N_) return;
  int b = t / N_, n = t % N_;
  const float* xb = x + (size_t)b * 3 * N_;
  float cx = xb[n], cy = xb[N_ + n], cz = xb[2 * N_ + n];
  float bd[KNN]; int bi[KNN];
#pragma unroll
  for (int j = 0; j < KNN; ++j) { bd[j] = FLT_MAX; bi[j] = 0; }
  float worst = FLT_MAX; int wpos = 0;
  for (int m = 0; m < N_; ++m) {
    float dx = xb[m] - cx, dy = xb[N_ + m] - cy, dz = xb[2 * N_ + m] - cz;
    float d = dx * dx + dy * dy + dz * dz;
    if (d < worst) {
      bd[wpos] = d; bi[wpos] = m;
      worst = bd[0]; wpos = 0;
#pragma unroll
      for (int j = 1; j < KNN; ++j)
        if (bd[j] > worst) { worst = bd[j]; wpos = j; }
    }
  }
#pragma unroll
  for (int j = 0; j < KNN; ++j) idx[(size_t)t * KNN + j] = bi[j];
}

// --- stage-1 BN statistics over (B,N,KNN): per-channel sum / sumsq ---------
// grid*block == B*N*KNN exactly (full waves required for shuffles).
__global__ void stats1_kernel(const float* __restrict__ x, const int* __restrict__ idx,
                              const float* __restrict__ Wpf,
                              float* __restrict__ gsum, float* __restrict__ gsq) {
  __shared__ float ssum[C1], ssq[C1];
  int tid = threadIdx.x;
  if (tid < C1) { ssum[tid] = 0.f; ssq[tid] = 0.f; }
  __syncthreads();
  int t = blockIdx.x * blockDim.x + tid;
  int b = t / (N_ * KNN);
  int r = t - b * (N_ * KNN);
  int n = r / KNN;
  int nb = idx[t];
  const float* xb = x + (size_t)b * 3 * N_;
  float f[3][3];
  load_feat(xb, n, nb, f);
  float nrm[C1];
#pragma unroll
  for (int c = 0; c < C1; ++c) {
    float px = 0.f, py = 0.f, pz = 0.f;
#pragma unroll
    for (int j = 0; j < 3; ++j) {
      float w = Wpf[c * 3 + j];
      px += w * f[j][0]; py += w * f[j][1]; pz += w * f[j][2];
    }
    nrm[c] = sqrtf(px * px + py * py + pz * pz) + EPS_;
  }
  int lane = tid & 31;
#pragma unroll
  for (int c = 0; c < C1; ++c) {
    float s = nrm[c], q = s * s;
#pragma unroll
    for (int off = 16; off > 0; off >>= 1) {
      s += __shfl_xor(s, off, 32);
      q += __shfl_xor(q, off, 32);
    }
    if (lane == 0) { atomicAdd(&ssum[c], s); atomicAdd(&ssq[c], q); }
  }
  __syncthreads();
  if (tid < C1) { atomicAdd(&gsum[tid], ssum[tid]); atomicAdd(&gsq[tid], ssq[tid]); }
}

// --- fused stage-1: feat -> VN-LReLU -> graph-pool(argmax over k) ----------
// one wave per (b,n); channel == lane; lanes >= C1 write zeros (pad rows).
__global__ void fused_pool1_kernel(const float* __restrict__ x, const int* __restrict__ idx,
                                   const float* __restrict__ Wpf, const float* __restrict__ Wpd,
                                   const float* __restrict__ Wg,
                                   const float* __restrict__ gsum, const float* __restrict__ gsq,
                                   float* __restrict__ hpool) {
  int wavesPerBlk = blockDim.x >> 5;
  int gw = blockIdx.x * wavesPerBlk + (threadIdx.x >> 5);
  if (gw >= B_ * N_) return;
  int lane = threadIdx.x & 31;
  int b = gw / N_, n = gw % N_;
  bool active = lane < C1;
  int c = active ? lane : (C1 - 1);
  const float cnt = (float)B_ * (float)N_ * (float)KNN;
  float mean = gsum[c] / cnt;
  float var  = gsq[c] / cnt - mean * mean;
  float istd = rsqrtf(var + BNEPS_);
  float wf0 = Wpf[c * 3 + 0], wf1 = Wpf[c * 3 + 1], wf2 = Wpf[c * 3 + 2];
  float wd0 = Wpd[c * 3 + 0], wd1 = Wpd[c * 3 + 1], wd2 = Wpd[c * 3 + 2];
  const float* xb = x + (size_t)b * 3 * N_;
  float bestdot = -FLT_MAX, bx = 0.f, by = 0.f, bz = 0.f;
  for (int k = 0; k < KNN; ++k) {
    int nb = idx[(size_t)gw * KNN + k];
    float f[3][3];
    load_feat(xb, n, nb, f);
    float px = wf0 * f[0][0] + wf1 * f[1][0] + wf2 * f[2][0];
    float py = wf0 * f[0][1] + wf1 * f[1][1] + wf2 * f[2][1];
    float pz = wf0 * f[0][2] + wf1 * f[1][2] + wf2 * f[2][2];
    float nrm = sqrtf(px * px + py * py + pz * pz) + EPS_;
    float fac = (nrm - mean) * istd / nrm;
    px *= fac; py *= fac; pz *= fac;
    float dx = wd0 * f[0][0] + wd1 * f[1][0] + wd2 * f[2][0];
    float dy = wd0 * f[0][1] + wd1 * f[1][1] + wd2 * f[2][1];
    float dz = wd0 * f[0][2] + wd1 * f[1][2] + wd2 * f[2][2];
    float dot = px * dx + py * dy + pz * dz;
    float dsq = dx * dx + dy * dy + dz * dz;
    if (dot < 0.f) { float s = dot / (dsq + EPS_); px -= s * dx; py -= s * dy; pz -= s * dz; }
    // graph-pool direction: dp_c = sum_c' Wg[c][c'] * h_c'  (cross-lane)
    float dpx = 0.f, dpy = 0.f, dpz = 0.f;
#pragma unroll
    for (int c2 = 0; c2 < C1; ++c2) {
      float hx = __shfl(px, c2, 32);
      float hy = __shfl(py, c2, 32);
      float hz = __shfl(pz, c2, 32);
      float wg = Wg[c * C1 + c2];
      dpx += wg * hx; dpy += wg * hy; dpz += wg * hz;
    }
    float dp = px * dpx + py * dpy + pz * dpz;
    if (active && dp > bestdot) { bestdot = dp; bx = px; by = py; bz = pz; }
  }
  if (!active) { bx = 0.f; by = 0.f; bz = 0.f; }   // pad rows -> exact zeros
  size_t base = (((size_t)b * Cp1 + lane) * 3) * N_ + n;
  hpool[base] = bx; hpool[base + N_] = by; hpool[base + 2 * N_] = bz;
}

// --- WMMA GEMM (generic, 1 tile/wave): Out[b] = Wp * X[b] ------------------
// Mp mult of 16, Kp mult of 4. Used only for layer 2 (Mp=48).
__global__ void wmma_gemm_kernel(const float* __restrict__ Wp, const float* __restrict__ X,
                                 float* __restrict__ Out, int Mp, int Kp, int Lc) {
  int lane = threadIdx.x & 31;
  int ltile = blockIdx.x * (blockDim.x >> 5) + (threadIdx.x >> 5);
  int half = lane >> 4, l16 = lane & 15;
  int l0 = (ltile << 4) + l16;
  int m0 = blockIdx.y << 4;
  int b  = blockIdx.z;
  const float* Arow = Wp + (size_t)(m0 + l16) * Kp + 2 * half;      // 8B aligned
  const float* Bcol = X + (size_t)b * Kp * Lc + (size_t)(2 * half) * Lc + l0;
  v8f acc = {};
#pragma unroll 2
  for (int k0 = 0; k0 < Kp; k0 += 4) {
    v2f a = *(const v2f*)(Arow + k0);
    v2f bb;
    bb.x = Bcol[(size_t)k0 * Lc];
    bb.y = Bcol[(size_t)k0 * Lc + Lc];
    acc = __builtin_amdgcn_wmma_f32_16x16x4_f32(false, a, false, bb,
                                                (short)0, acc, false, false);
  }
  float* Ob = Out + (size_t)b * Mp * Lc + l0;
#pragma unroll
  for (int v = 0; v < 8; ++v)
    Ob[(size_t)(m0 + v + 8 * half) * Lc] = acc[v];
}

// --- WMMA GEMM, 2x2 register-blocked: 32x32 output per wave ----------------
// Mp mult of 32, Kp mult of 4. Per K-step: 2x b64 A + 4x b32 B + 4 WMMAs.
__global__ void wmma_gemm_2x2_kernel(const float* __restrict__ Wp, const float* __restrict__ X,
                                     float* __restrict__ Out, int Mp, int Kp, int Lc) {
  int lane = threadIdx.x & 31;
  int lpair = blockIdx.x * (blockDim.x >> 5) + (threadIdx.x >> 5);
  int half = lane >> 4, l16 = lane & 15;
  int l0 = (lpair << 5) + l16;       // first of two adjacent 16-col tiles
  int m0 = blockIdx.y << 5;          // first of two adjacent 16-row tiles
  int b  = blockIdx.z;
  const float* Arow0 = Wp + (size_t)(m0 + l16) * Kp + 2 * half;
  const float* Arow1 = Arow0 + (size_t)16 * Kp;
  const float* Bcol0 = X + (size_t)b * Kp * Lc + (size_t)(2 * half) * Lc + l0;
  const float* Bcol1 = Bcol0 + 16;
  v8f acc00 = {}, acc01 = {}, acc10 = {}, acc11 = {};
#pragma unroll 2
  for (int k0 = 0; k0 < Kp; k0 += 4) {
    v2f a0 = *(const v2f*)(Arow0 + k0);
    v2f a1 = *(const v2f*)(Arow1 + k0);
    v2f b0, b1;
    b0.x = Bcol0[(size_t)k0 * Lc];
    b0.y = Bcol0[(size_t)k0 * Lc + Lc];
    b1.x = Bcol1[(size_t)k0 * Lc];
    b1.y = Bcol1[(size_t)k0 * Lc + Lc];
    acc00 = __builtin_amdgcn_wmma_f32_16x16x4_f32(false, a0, false, b0,
                                                  (short)0, acc00, false, false);
    acc01 = __builtin_amdgcn_wmma_f32_16x16x4_f32(false, a0, false, b1,
                                                  (short)0, acc01, false, false);
    acc10 = __builtin_amdgcn_wmma_f32_16x16x4_f32(false, a1, false, b0,
                                                  (short)0, acc10, false, false);
    acc11 = __builtin_amdgcn_wmma_f32_16x16x4_f32(false, a1, false, b1,
                                                  (short)0, acc11, false, false);
  }
  float* Ob = Out + (size_t)b * Mp * Lc + l0;
#pragma unroll
  for (int v = 0; v < 8; ++v) {
    size_t r0 = (size_t)(m0 + v + 8 * half) * Lc;
    size_t r1 = (size_t)(m0 + 16 + v + 8 * half) * Lc;
    Ob[r0]      = acc00[v];
    Ob[r0 + 16] = acc01[v];
    Ob[r1]      = acc10[v];
    Ob[r1 + 16] = acc11[v];
  }
}

// --- BN stats over (B,N); p laid out (B,Cp,3,N): grid (N/256, C, B) --------
__global__ void bn_stats_kernel(const float* __restrict__ p, float* __restrict__ gsum,
                                float* __restrict__ gsq, int Cp) {
  __shared__ float ss[256], sq[256];
  int n = blockIdx.x * blockDim.x + threadIdx.x;
  int c = blockIdx.y, b = blockIdx.z;
  size_t base = (((size_t)b * Cp + c) * 3) * N_ + n;
  float x0 = p[base], x1 = p[base + N_], x2 = p[base + 2 * N_];
  float nrm = sqrtf(x0 * x0 + x1 * x1 + x2 * x2) + EPS_;
  ss[threadIdx.x] = nrm; sq[threadIdx.x] = nrm * nrm;
  __syncthreads();
  for (int s = blockDim.x >> 1; s > 0; s >>= 1) {
    if ((int)threadIdx.x < s) {
      ss[threadIdx.x] += ss[threadIdx.x + s];
      sq[threadIdx.x] += sq[threadIdx.x + s];
    }
    __syncthreads();
  }
  if (threadIdx.x == 0) { atomicAdd(&gsum[c], ss[0]); atomicAdd(&gsq[c], sq[0]); }
}

// --- apply BN + VN-LeakyReLU(0); in-place safe; pad rows untouched ---------
__global__ void bn_lrelu_kernel(const float* __restrict__ p, const float* __restrict__ d,
                                float* __restrict__ h, const float* __restrict__ gsum,
                                const float* __restrict__ gsq, int C, int Cp, float cnt) {
  int t = blockIdx.x * blockDim.x + threadIdx.x;
  if (t >= B_ * C * N_) return;
  int b = t / (C * N_);
  int r = t - b * C * N_;
  int c = r / N_, n = r - c * N_;
  size_t base = (((size_t)b * Cp + c) * 3) * N_ + n;
  float px = p[base], py = p[base + N_], pz = p[base + 2 * N_];
  float dx = d[base], dy = d[base + N_], dz = d[base + 2 * N_];
  float nrm = sqrtf(px * px + py * py + pz * pz) + EPS_;
  float mean = gsum[c] / cnt;
  float var  = gsq[c] / cnt - mean * mean;
  float fac = (nrm - mean) * rsqrtf(var + BNEPS_) / nrm;
  px *= fac; py *= fac; pz *= fac;
  float dot = px * dx + py * dy + pz * dz;
  float dsq = dx * dx + dy * dy + dz * dz;
  if (dot < 0.f) { float s = dot / (dsq + EPS_); px -= s * dx; py -= s * dy; pz -= s * dz; }
  h[base] = px; h[base + N_] = py; h[base + 2 * N_] = pz;
}

// --- max-pool over N: one wave per (b,c), argmax of <h,d> ------------------
__global__ void pool_n_kernel(const float* __restrict__ h, const float* __restrict__ d,
                              float* __restrict__ pooled, int C, int Cp) {
  int wavesPerBlk = blockDim.x >> 5;
  int gw = blockIdx.x * wavesPerBlk + (threadIdx.x >> 5);
  if (gw >= B_ * C) return;
  int lane = threadIdx.x & 31;
  int b = gw / C, c = gw % C;
  size_t base = (((size_t)b * Cp + c) * 3) * N_;
  float bestdot = -FLT_MAX; int bestn = 0;
  for (int n = lane; n < N_; n += 32) {
    float dot = h[base + n] * d[base + n]
              + h[base + N_ + n] * d[base + N_ + n]
              + h[base + 2 * N_ + n] * d[base + 2 * N_ + n];
    if (dot > bestdot || (dot == bestdot && n < bestn)) { bestdot = dot; bestn = n; }
  }
#pragma unroll
  for (int off = 16; off > 0; off >>= 1) {
    float od = __shfl_xor(bestdot, off, 32);
    int   on = __shfl_xor(bestn, off, 32);
    if (od > bestdot || (od == bestdot && on < bestn)) { bestdot = od; bestn = on; }
  }
  if (lane == 0) {
    pooled[((size_t)b * C + c) * 3 + 0] = h[base + bestn];
    pooled[((size_t)b * C + c) * 3 + 1] = h[base + N_ + bestn];
    pooled[((size_t)b * C + c) * 3 + 2] = h[base + 2 * N_ + bestn];
  }
}

// --- tiny FC head ----------------------------------------------------------
__global__ void head_lin_kernel(const float* __restrict__ W, const float* __restrict__ in,
                                float* __restrict__ out, int Cout, int Cin) {
  int t = blockIdx.x * blockDim.x + threadIdx.x;
  if (t >= B_ * Cout * 3) return;
  int b = t / (Cout * 3);
  int r = t - b * Cout * 3;
  int co = r / 3, v = r % 3;
  float acc = 0.f;
  for (int ci = 0; ci < Cin; ++ci)
    acc += W[co * Cin + ci] * in[((size_t)b * Cin + ci) * 3 + v];
  out[((size_t)b * Cout + co) * 3 + v] = acc;
}

__global__ void head_stats_kernel(const float* __restrict__ p, float* __restrict__ mean,
                                  float* __restrict__ var, int C) {
  int c = blockIdx.x * blockDim.x + threadIdx.x;
  if (c >= C) return;
  float s = 0.f, q = 0.f;
  for (int b = 0; b < B_; ++b) {
    size_t base = ((size_t)b * C + c) * 3;
    float x0 = p[base], x1 = p[base + 1], x2 = p[base + 2];
    float nrm = sqrtf(x0 * x0 + x1 * x1 + x2 * x2) + EPS_;
    s += nrm; q += nrm * nrm;
  }
  float m = s / (float)B_;
  mean[c] = m; var[c] = q / (float)B_ - m * m;
}

__global__ void head_lrelu_kernel(const float* __restrict__ p, const float* __restrict__ d,
                                  float* __restrict__ h, const float* __restrict__ mean,
                                  const float* __restrict__ var, int C) {
  int t = blockIdx.x * blockDim.x + threadIdx.x;
  if (t >= B_ * C) return;
  int b = t / C, c = t % C;
  size_t base = ((size_t)b * C + c) * 3;
  float px = p[base], py = p[base + 1], pz = p[base + 2];
  float dx = d[base], dy = d[base + 1], dz = d[base + 2];
  float nrm = sqrtf(px * px + py * py + pz * pz) + EPS_;
  float fac = (nrm - mean[c]) * rsqrtf(var[c] + BNEPS_) / nrm;
  px *= fac; py *= fac; pz *= fac;
  float dot = px * dx + py * dy + pz * dz;
  float dsq = dx * dx + dy * dy + dz * dz;
  if (dot < 0.f) { float s = dot / (dsq + EPS_); px -= s * dx; py -= s * dy; pz -= s * dz; }
  h[base] = px; h[base + 1] = py; h[base + 2] = pz;
}

// ---------------------------------------------------------------------------
extern "C" void kernel_launch(void* const* d_in, const int* in_sizes, int n_in,
                              void* d_out, int out_size, void* d_ws, size_t ws_size,
                              hipStream_t stream) {
  (void)in_sizes; (void)n_in; (void)out_size; (void)ws_size;
  const float* x    = (const float*)d_in[0];
  const float* Wpf  = (const float*)d_in[1];
  const float* Wpd  = (const float*)d_in[2];
  const float* Wg   = (const float*)d_in[3];
  const float* W1f  = (const float*)d_in[4];
  const float* W1d  = (const float*)d_in[5];
  const float* W2f  = (const float*)d_in[6];
  const float* W2d  = (const float*)d_in[7];
  const float* W3f  = (const float*)d_in[8];
  const float* W3d  = (const float*)d_in[9];
  const float* Wpl  = (const float*)d_in[10];
  const float* Wf1f = (const float*)d_in[11];
  const float* Wf1d = (const float*)d_in[12];
  const float* Wf2f = (const float*)d_in[13];
  const float* Wf2d = (const float*)d_in[14];
  const float* Wf3  = (const float*)d_in[15];
  float* out = (float*)d_out;

  // workspace carving (256B aligned)
  char* base = (char*)d_ws;
  size_t off = 0;
  auto carve = [&](size_t bytes) -> void* {
    void* p = base + off;
    off = (off + bytes + 255) & ~(size_t)255;
    return p;
  };
  int*   idxb   = (int*)  carve(sizeof(int)   * (size_t)B_ * N_ * KNN);
  float* bufA   = (float*)carve(sizeof(float) * (size_t)B_ * Cp3 * 3 * N_); // 69MB
  float* bufB   = (float*)carve(sizeof(float) * (size_t)B_ * Cp3 * 3 * N_); // 69MB
  float* bufC   = (float*)carve(sizeof(float) * (size_t)B_ * Cp2 * 3 * N_);
  float* bufH1  = (float*)carve(sizeof(float) * (size_t)B_ * Cp1 * 3 * N_);
  float* pooled = (float*)carve(sizeof(float) * (size_t)B_ * C3 * 3);
  float* hP1    = (float*)carve(sizeof(float) * (size_t)B_ * F1 * 3);
  float* hD1    = (float*)carve(sizeof(float) * (size_t)B_ * F1 * 3);
  float* hH1    = (float*)carve(sizeof(float) * (size_t)B_ * F1 * 3);
  float* hP2    = (float*)carve(sizeof(float) * (size_t)B_ * F2 * 3);
  float* hD2    = (float*)carve(sizeof(float) * (size_t)B_ * F2 * 3);
  float* hH2    = (float*)carve(sizeof(float) * (size_t)B_ * F2 * 3);
  // zero-padded weights
  float* W1fp  = (float*)carve(sizeof(float) * Cp1 * Cp1);
  float* W1dp  = (float*)carve(sizeof(float) * Cp1 * Cp1);
  float* W2fp  = (float*)carve(sizeof(float) * Cp2 * Cp1);
  float* W2dp  = (float*)carve(sizeof(float) * Cp2 * Cp1);
  float* W3fp  = (float*)carve(sizeof(float) * Cp3 * Cp2);
  float* W3dp  = (float*)carve(sizeof(float) * Cp3 * Cp2);
  float* Wplp  = (float*)carve(sizeof(float) * Cp3 * Cp3);
  float* stSum = (float*)carve(sizeof(float) * 1024);   // [0..511]=sum, [512..]=sumsq
  float* stSq  = stSum + 512;

  auto padw = [&](const float* W, float* Wp, int M, int K, int Mp, int Kp) {
    pad_weight_kernel<<<(Mp * Kp + 255) / 256, 256, 0, stream>>>(W, Wp, M, K, Mp, Kp);
  };
  auto gemm = [&](const float* Wp, const float* X, float* O, int Mp, int Kp) {
    if ((Mp & 31) == 0) {                              // 2x2 blocked path
      dim3 grid((LCOLS / 32) / 8, Mp / 32, B_);        // 24 x (Mp/32) x 8
      wmma_gemm_2x2_kernel<<<grid, 256, 0, stream>>>(Wp, X, O, Mp, Kp, LCOLS);
    } else {                                           // generic path (layer 2)
      dim3 grid((LCOLS / 16) / 8, Mp / 16, B_);        // 48 x (Mp/16) x 8
      wmma_gemm_kernel<<<grid, 256, 0, stream>>>(Wp, X, O, Mp, Kp, LCOLS);
    }
  };

  // 0) pad all GEMM weights (tiny, once per call)
  padw(W1f, W1fp, C1, C1, Cp1, Cp1);
  padw(W1d, W1dp, C1, C1, Cp1, Cp1);
  padw(W2f, W2fp, C2, C1, Cp2, Cp1);
  padw(W2d, W2dp, C2, C1, Cp2, Cp1);
  padw(W3f, W3fp, C3, C2, Cp3, Cp2);
  padw(W3d, W3dp, C3, C2, Cp3, Cp2);
  padw(Wpl, Wplp, C3, C3, Cp3, Cp3);

  // 1) kNN graph
  knn_kernel<<<(B_ * N_) / 256, 256, 0, stream>>>(x, idxb);

  // 2) stage 1: stats over (B,N,K), then fused VN-LReLU + graph pool over k
  hipMemsetAsync(stSum, 0, sizeof(float) * 1024, stream);
  stats1_kernel<<<(B_ * N_ * KNN) / 256, 256, 0, stream>>>(x, idxb, Wpf, stSum, stSq);
  fused_pool1_kernel<<<(B_ * N_) / 8, 256, 0, stream>>>(x, idxb, Wpf, Wpd, Wg,
                                                        stSum, stSq, bufH1);

  // 3) layer 1: 21 -> 21   (padded 32x32, blocked kernel)
  gemm(W1fp, bufH1, bufA, Cp1, Cp1);
  gemm(W1dp, bufH1, bufB, Cp1, Cp1);
  hipMemsetAsync(stSum, 0, sizeof(float) * 1024, stream);
  bn_stats_kernel<<<dim3(N_ / 256, C1, B_), 256, 0, stream>>>(bufA, stSum, stSq, Cp1);
  bn_lrelu_kernel<<<(B_ * C1 * N_) / 256, 256, 0, stream>>>(bufA, bufB, bufA,
                                                            stSum, stSq, C1, Cp1,
                                                            (float)(B_ * N_));

  // 4) layer 2: 21 -> 42   (padded 48x32, generic kernel)
  gemm(W2fp, bufA, bufC, Cp2, Cp1);
  gemm(W2dp, bufA, bufB, Cp2, Cp1);
  hipMemsetAsync(stSum, 0, sizeof(float) * 1024, stream);
  bn_stats_kernel<<<dim3(N_ / 256, C2, B_), 256, 0, stream>>>(bufC, stSum, stSq, Cp2);
  bn_lrelu_kernel<<<(B_ * C2 * N_) / 256, 256, 0, stream>>>(bufC, bufB, bufC,
                                                            stSum, stSq, C2, Cp2,
                                                            (float)(B_ * N_));

  // 5) layer 3: 42 -> 341  (padded 352x48, blocked kernel)
  gemm(W3fp, bufC, bufA, Cp3, Cp2);
  gemm(W3dp, bufC, bufB, Cp3, Cp2);
  hipMemsetAsync(stSum, 0, sizeof(float) * 1024, stream);
  bn_stats_kernel<<<dim3(N_ / 256, C3, B_), 256, 0, stream>>>(bufA, stSum, stSq, Cp3);
  bn_lrelu_kernel<<<(B_ * C3 * N_) / 256, 256, 0, stream>>>(bufA, bufB, bufA,
                                                            stSum, stSq, C3, Cp3,
                                                            (float)(B_ * N_));

  // 6) global max-pool over N (direction GEMM padded 352x352, blocked kernel)
  gemm(Wplp, bufA, bufB, Cp3, Cp3);
  pool_n_kernel<<<(B_ * C3) / 8, 256, 0, stream>>>(bufA, bufB, pooled, C3, Cp3);

  // 7) FC head (tiny): 341 -> 170 -> 85 -> 3
  head_lin_kernel<<<(B_ * F1 * 3 + 255) / 256, 256, 0, stream>>>(Wf1f, pooled, hP1, F1, C3);
  head_lin_kernel<<<(B_ * F1 * 3 + 255) / 256, 256, 0, stream>>>(Wf1d, pooled, hD1, F1, C3);
  head_stats_kernel<<<1, 256, 0, stream>>>(hP1, stSum, stSq, F1);
  head_lrelu_kernel<<<(B_ * F1 + 255) / 256, 256, 0, stream>>>(hP1, hD1, hH1, stSum, stSq, F1);

  head_lin_kernel<<<(B_ * F2 * 3 + 255) / 256, 256, 0, stream>>>(Wf2f, hH1, hP2, F2, F1);
  head_lin_kernel<<<(B_ * F2 * 3 + 255) / 256, 256, 0, stream>>>(Wf2d, hH1, hD2, F2, F1);
  head_stats_kernel<<<1, 256, 0, stream>>>(hP2, stSum, stSq, F2);
  head_lrelu_kernel<<<(B_ * F2 + 255) / 256, 256, 0, stream>>>(hP2, hD2, hH2, stSum, stSq, F2);

  head_lin_kernel<<<1, 256, 0, stream>>>(Wf3, hH2, out, 3, F2);   // (B,3,3) output
}